// TGNMemory_63496796504572
// MI455X (gfx1250) — compile-verified
//
#include <hip/hip_runtime.h>
#include <math.h>

typedef __attribute__((ext_vector_type(2))) float v2f;
typedef __attribute__((ext_vector_type(8))) float v8f;

#define MEMD 128
#define MSGD 512

__device__ __forceinline__ v8f wmma4(v2f a, v2f b, v8f c) {
    // D = A(16x4 f32) * B(4x16 f32) + C(16x16 f32), full fp32 precision
    return __builtin_amdgcn_wmma_f32_16x16x4_f32(
        /*neg_a=*/false, a, /*neg_b=*/false, b,
        /*c_mod=*/(short)0, c, /*reuse_a=*/false, /*reuse_b=*/false);
}

// ---------------- Kernel A: zero workspace + last_update output ----------------
__global__ __launch_bounds__(256) void tgn_zero(float* __restrict__ sums,
                                                float* __restrict__ cnt,
                                                int* __restrict__ lu,
                                                long nsum, int nnodes) {
    long i = (long)blockIdx.x * blockDim.x + threadIdx.x;
    long stride = (long)gridDim.x * blockDim.x;
    for (long j = i; j < nsum; j += stride) sums[j] = 0.0f;
    for (long j = i; j < nnodes; j += stride) { cnt[j] = 0.0f; lu[j] = 0; }
}

// ---------------- Kernel B: per-event message scatter (atomic mean-aggregate) --
__global__ __launch_bounds__(128) void tgn_scatter(
    const float* __restrict__ memory, const int* __restrict__ last_update,
    const int* __restrict__ src, const int* __restrict__ dst,
    const int* __restrict__ t, const float* __restrict__ raw,
    const float* __restrict__ tw, const float* __restrict__ tb,
    float* __restrict__ sums, float* __restrict__ cnt, int* __restrict__ lu_out,
    int E) {
    int e = blockIdx.x;
    if (e >= E) return;
    int f = threadIdx.x;                 // 0..127
    int s = src[e], d = dst[e], te = t[e];

    float ms = memory[(long)s * MEMD + f];
    float md = memory[(long)d * MEMD + f];
    float rw = raw[(long)e * MEMD + f];
    float w = tw[f], b = tb[f];
    float trs = (float)(te - last_update[s]);
    float trd = (float)(te - last_update[d]);
    float tes = cosf(trs * w + b);       // time encoding, src direction
    float ted = cosf(trd * w + b);       // time encoding, dst direction

    float* Ss = sums + (long)s * MSGD;
    float* Sd = sums + (long)d * MSGD;
    // msg_s -> node src: [mem[s], mem[d], raw, tenc(lu[s])]
    unsafeAtomicAdd(Ss + f, ms);
    unsafeAtomicAdd(Ss + 128 + f, md);
    unsafeAtomicAdd(Ss + 256 + f, rw);
    unsafeAtomicAdd(Ss + 384 + f, tes);
    // msg_d -> node dst: [mem[d], mem[s], raw, tenc(lu[d])]
    unsafeAtomicAdd(Sd + f, md);
    unsafeAtomicAdd(Sd + 128 + f, ms);
    unsafeAtomicAdd(Sd + 256 + f, rw);
    unsafeAtomicAdd(Sd + 384 + f, ted);

    if (f == 0) {
        unsafeAtomicAdd(&cnt[s], 1.0f);
        unsafeAtomicAdd(&cnt[d], 1.0f);
        atomicMax(&lu_out[s], te);
        atomicMax(&lu_out[d], te);
    }
}

// ---------------- Kernel C: fused (aggr@Wih^T, mem@Whh^T) WMMA GEMM + GRU ------
__global__ __launch_bounds__(256) void tgn_gru_wmma(
    const float* __restrict__ sums, const float* __restrict__ cnt,
    const float* __restrict__ memory,
    const float* __restrict__ Wih, const float* __restrict__ Whh,
    const float* __restrict__ bih, const float* __restrict__ bhh,
    float* __restrict__ out, int nnodes) {
    __shared__ float sA[16][516];   // aggr tile, padded (bank-conflict-free frags)
    __shared__ float sM[16][132];   // memory tile, padded
    __shared__ float sInv[16];

    const int tid = threadIdx.x;
    const int node0 = blockIdx.x * 16;

    if (tid < 16) {
        int n = node0 + tid; if (n >= nnodes) n = nnodes - 1;
        sInv[tid] = 1.0f / fmaxf(cnt[n], 1.0f);
    }
    __syncthreads();
    for (int i = tid; i < 16 * 512; i += 256) {
        int r = i >> 9, c = i & 511;
        int n = node0 + r; if (n >= nnodes) n = nnodes - 1;
        sA[r][c] = sums[(long)n * 512 + c] * sInv[r];
    }
    for (int i = tid; i < 16 * 128; i += 256) {
        int r = i >> 7, c = i & 127;
        int n = node0 + r; if (n >= nnodes) n = nnodes - 1;
        sM[r][c] = memory[(long)n * 128 + c];
    }
    __syncthreads();

    const int wave = tid >> 5, lane = tid & 31;
    const int half = lane >> 4;          // 0: k={0,1}, M/N 0..15 ; 1: k={2,3}
    const int l16 = lane & 15;
    const int koff = half << 1;
    const int f0 = wave << 4;            // feature tile base (wave w -> cols 16w..)
    const int col = f0 + l16;            // 0..127

    // Weight rows for the three gates at this lane's output column
    const float* Wr = Wih + (long)(col) * 512;
    const float* Wz = Wih + (long)(128 + col) * 512;
    const float* Wn = Wih + (long)(256 + col) * 512;
    const float* Hr = Whh + (long)(col) * 128;
    const float* Hz = Whh + (long)(128 + col) * 128;
    const float* Hn = Whh + (long)(256 + col) * 128;

    v8f gir = {}, giz = {}, gin = {};
    v8f ghr = {}, ghz = {}, ghn = {};

    // gi = aggr @ Wih^T : K = 512 -> 128 WMMA k-steps, 3 gates each
#pragma unroll 4
    for (int kk = 0; kk < 128; ++kk) {
        v2f a = *(const v2f*)&sA[l16][(kk << 2) + koff];
        v2f b0 = *(const v2f*)(Wr + (kk << 2) + koff);
        v2f b1 = *(const v2f*)(Wz + (kk << 2) + koff);
        v2f b2 = *(const v2f*)(Wn + (kk << 2) + koff);
        gir = wmma4(a, b0, gir);
        giz = wmma4(a, b1, giz);
        gin = wmma4(a, b2, gin);
    }
    // gh = memory @ Whh^T : K = 128 -> 32 WMMA k-steps
#pragma unroll 4
    for (int kk = 0; kk < 32; ++kk) {
        v2f a = *(const v2f*)&sM[l16][(kk << 2) + koff];
        v2f b0 = *(const v2f*)(Hr + (kk << 2) + koff);
        v2f b1 = *(const v2f*)(Hz + (kk << 2) + koff);
        v2f b2 = *(const v2f*)(Hn + (kk << 2) + koff);
        ghr = wmma4(a, b0, ghr);
        ghz = wmma4(a, b1, ghz);
        ghn = wmma4(a, b2, ghn);
    }

    const float bir = bih[col], biz = bih[128 + col], bin = bih[256 + col];
    const float bhr = bhh[col], bhz = bhh[128 + col], bhn = bhh[256 + col];

    // GRU elementwise in the WMMA C/D layout: VGPR v -> row M, lane -> col N
#pragma unroll
    for (int v = 0; v < 8; ++v) {
        int row = half ? (v + 8) : v;
        float ir = gir[v] + bir, iz = giz[v] + biz, in_ = gin[v] + bin;
        float hr = ghr[v] + bhr, hz = ghz[v] + bhz, hn = ghn[v] + bhn;
        float r = 1.0f / (1.0f + expf(-(ir + hr)));
        float z = 1.0f / (1.0f + expf(-(iz + hz)));
        float n = tanhf(in_ + r * hn);
        float m = sM[row][col];
        int node = node0 + row;
        if (node < nnodes)
            out[(long)node * 128 + col] = (1.0f - z) * n + z * m;
    }
}

extern "C" void kernel_launch(void* const* d_in, const int* in_sizes, int n_in,
                              void* d_out, int out_size, void* d_ws, size_t ws_size,
                              hipStream_t stream) {
    const float* memory      = (const float*)d_in[0];
    const int*   last_update = (const int*)d_in[1];
    const int*   src         = (const int*)d_in[2];
    const int*   dst         = (const int*)d_in[3];
    const int*   t           = (const int*)d_in[4];
    const float* raw         = (const float*)d_in[5];
    const float* tw          = (const float*)d_in[6];
    const float* tb          = (const float*)d_in[7];
    const float* Wih         = (const float*)d_in[8];
    const float* Whh         = (const float*)d_in[9];
    const float* bih         = (const float*)d_in[10];
    const float* bhh         = (const float*)d_in[11];

    const int N = in_sizes[1];   // 200000 nodes
    const int E = in_sizes[2];   // 150000 events

    float* out_mem = (float*)d_out;
    int*   out_lu  = (int*)((float*)d_out + (long)N * MEMD);  // int32 bits in tail

    float* sums = (float*)d_ws;                // [N, 512] f32
    float* cnt  = sums + (long)N * MSGD;       // [N] f32
    long nsum = (long)N * MSGD;

    tgn_zero<<<4096, 256, 0, stream>>>(sums, cnt, out_lu, nsum, N);
    tgn_scatter<<<E, 128, 0, stream>>>(memory, last_update, src, dst, t, raw,
                                       tw, tb, sums, cnt, out_lu, E);
    int nblk = (N + 15) / 16;
    tgn_gru_wmma<<<nblk, 256, 0, stream>>>(sums, cnt, memory, Wih, Whh, bih, bhh,
                                           out_mem, N);
}